// GaiaModel_67817533604060
// MI455X (gfx1250) — compile-verified
//
#include <hip/hip_runtime.h>
#include <hip/hip_bf16.h>
#include <math.h>

#define KNN_K 8
#define LAT_N 91
#define LON_N 180
#define GRID  (LAT_N * LON_N)    // 16380
#define NODES 10242
#define CH    64
#define BATCH 4
#define ROWS  (BATCH * GRID)     // 65520
#define ROW_TILES (ROWS / 16)    // 4095 (exact)

typedef __attribute__((ext_vector_type(2))) float v2f;
typedef __attribute__((ext_vector_type(8))) float v8f;

// ---------------------------------------------------------------------------
// Kernel 1: fused KNN (top-8 by distance, lowest-index tiebreak) + softmax.
// One wave32 per grid point. Streaming register top-8 per lane (branchless
// sorted-insert, fully unrolled -> stays in VGPRs), LDS merge of 32x8
// candidates, lane 0 finalizes + softmax. Avoids materializing the
// 16380x10242 distance matrix entirely.
// ---------------------------------------------------------------------------
__global__ __launch_bounds__(32) void knn_softmax_kernel(
    const float* __restrict__ mesh_vertices,   // [NODES][3]
    const float* __restrict__ lat,             // [LAT_N]
    const float* __restrict__ lon,             // [LON_N]
    int*   __restrict__ knn_idx,               // [GRID][KNN_K]
    float* __restrict__ knn_w)                 // [GRID][KNN_K]
{
    const int g    = blockIdx.x;
    const int lane = threadIdx.x;
    const int la   = g / LON_N;
    const int lo   = g - la * LON_N;

    const float latv = lat[la];
    const float lonv = lon[lo];
    const float cl = cosf(latv);
    const float gx = cl * cosf(lonv);
    const float gy = cl * sinf(lonv);
    const float gz = sinf(latv);
    const float g2 = gx * gx + gy * gy + gz * gz;

    float bd[KNN_K];
    int   bi[KNN_K];
#pragma unroll
    for (int i = 0; i < KNN_K; ++i) { bd[i] = 3.0e38f; bi[i] = 0x7fffffff; }

    for (int v = lane; v < NODES; v += 32) {
        const float mx = mesh_vertices[3 * v + 0];
        const float my = mesh_vertices[3 * v + 1];
        const float mz = mesh_vertices[3 * v + 2];
        const float m2  = mx * mx + my * my + mz * mz;
        const float dot = gx * mx + gy * my + gz * mz;
        const float d2  = g2 + m2 - 2.0f * dot;
        float cd = sqrtf(fmaxf(d2, 1e-12f));
        int   ci = v;
        // branchless sorted insert (ascending (dist, idx))
#pragma unroll
        for (int j = 0; j < KNN_K; ++j) {
            const bool better = (cd < bd[j]) || ((cd == bd[j]) && (ci < bi[j]));
            const float td = bd[j]; const int ti = bi[j];
            if (better) { bd[j] = cd; bi[j] = ci; cd = td; ci = ti; }
        }
    }

    __shared__ float sd[32 * KNN_K];
    __shared__ int   si[32 * KNN_K];
#pragma unroll
    for (int i = 0; i < KNN_K; ++i) {
        sd[lane * KNN_K + i] = bd[i];
        si[lane * KNN_K + i] = bi[i];
    }
    __syncthreads();

    if (lane == 0) {
        float fd[KNN_K]; int fi[KNN_K];
#pragma unroll
        for (int i = 0; i < KNN_K; ++i) { fd[i] = 3.0e38f; fi[i] = 0x7fffffff; }
        for (int t = 0; t < 32 * KNN_K; ++t) {
            float cd = sd[t]; int ci = si[t];
#pragma unroll
            for (int j = 0; j < KNN_K; ++j) {
                const bool better = (cd < fd[j]) || ((cd == fd[j]) && (ci < fi[j]));
                const float td = fd[j]; const int ti = fi[j];
                if (better) { fd[j] = cd; fi[j] = ci; cd = td; ci = ti; }
            }
        }
        // softmax over -dist; fd sorted ascending => fd[0] is the max of -dist
        float w[KNN_K]; float s = 0.0f;
#pragma unroll
        for (int i = 0; i < KNN_K; ++i) { w[i] = expf(fd[0] - fd[i]); s += w[i]; }
        const float inv = 1.0f / s;
#pragma unroll
        for (int i = 0; i < KNN_K; ++i) {
            knn_idx[g * KNN_K + i] = fi[i];
            knn_w [g * KNN_K + i] = w[i] * inv;
        }
    }
}

// ---------------------------------------------------------------------------
// Kernel 2: gather + weighted interpolation + channel-mix GEMM via fp32 WMMA.
// One wave32 per 16-row tile of the flattened (B*GRID, 64) activation matrix.
// X(16x64) built in LDS; D(16x16) tiles = A(16x4) x B(4x16) accumulated over
// 16 K-steps with V_WMMA_F32_16X16X4_F32 (fp32 in/out: matches reference
// precision). B = W^T fragments read from an LDS copy of W. Output scattered
// into (B, C, lat, lon) layout (out idx = (b*64+c)*16380 + g).
// ---------------------------------------------------------------------------
__global__ __launch_bounds__(32) void interp_gemm_kernel(
    const float* __restrict__ mesh_output,   // [BATCH][NODES][CH]
    const float* __restrict__ W,             // [CH][CH]
    const float* __restrict__ bias,          // [CH]
    const int*   __restrict__ knn_idx,       // [GRID][KNN_K]
    const float* __restrict__ knn_w,         // [GRID][KNN_K]
    float* __restrict__ out)                 // [BATCH][CH][GRID]
{
    __shared__ float Xt[16][CH];     // interpolated activation tile
    __shared__ float Wl[CH * CH];    // W, row-major

    const int lane = threadIdx.x;
    const int tile = blockIdx.x;
    const int R0   = tile * 16;

    // stage W into LDS (4096 floats; 32 float4 per lane)
    {
        const float4* src = (const float4*)W;
        float4*       dst = (float4*)Wl;
#pragma unroll
        for (int i = 0; i < (CH * CH / 4) / 32; ++i)
            dst[lane + i * 32] = src[lane + i * 32];
    }

    // build Xt: each lane owns 2 channels across all 16 rows
    const int c0 = lane * 2;
    for (int r = 0; r < 16; ++r) {
        const int R = R0 + r;
        const int b = R / GRID;
        const int g = R - b * GRID;
        float acc0 = 0.0f, acc1 = 0.0f;
#pragma unroll
        for (int k = 0; k < KNN_K; ++k) {
            const int   id = knn_idx[g * KNN_K + k];
            const float wk = knn_w [g * KNN_K + k];
            const float* src = mesh_output + ((size_t)b * NODES + id) * CH + c0;
            acc0 = fmaf(wk, src[0], acc0);
            acc1 = fmaf(wk, src[1], acc1);
        }
        Xt[r][c0]     = acc0;
        Xt[r][c0 + 1] = acc1;
    }
    __syncthreads();

    // A-fragment layout (32-bit 16x4): lanes 0-15 -> M=lane, K={0,1};
    // lanes 16-31 -> M=lane-16, K={2,3}
    const int mrow = lane & 15;
    const int koff = (lane >> 4) * 2;

    v2f a[16];
#pragma unroll
    for (int t = 0; t < 16; ++t) {
        a[t].x = Xt[mrow][4 * t + koff];
        a[t].y = Xt[mrow][4 * t + koff + 1];
    }

#pragma unroll
    for (int ct = 0; ct < 4; ++ct) {
        const int c = ct * 16 + mrow;              // output channel (N)
        v8f acc = {0.f, 0.f, 0.f, 0.f, 0.f, 0.f, 0.f, 0.f};
#pragma unroll
        for (int t = 0; t < 16; ++t) {
            // B(4x16) fragment: B[j][n] = W[n][j]; lane n=mrow, rows j=4t+koff,+1
            v2f bf;
            const int j = 4 * t + koff;
            bf.x = Wl[c * CH + j];
            bf.y = Wl[c * CH + j + 1];
            acc = __builtin_amdgcn_wmma_f32_16x16x4_f32(
                false, a[t], false, bf, (short)0, acc, false, false);
        }
        const float bv = bias[c];
        // D layout: element v -> M = v + (lane>=16 ? 8 : 0), N = lane&15
#pragma unroll
        for (int v = 0; v < 8; ++v) {
            const int m = v + ((lane >> 4) << 3);
            const int R = R0 + m;
            const int b = R / GRID;
            const int g = R - b * GRID;
            out[((size_t)b * CH + c) * GRID + g] = acc[v] + bv;
        }
    }
}

extern "C" void kernel_launch(void* const* d_in, const int* in_sizes, int n_in,
                              void* d_out, int out_size, void* d_ws, size_t ws_size,
                              hipStream_t stream) {
    const float* mesh_output   = (const float*)d_in[0];  // [4][10242][64]
    const float* mesh_vertices = (const float*)d_in[1];  // [10242][3]
    const float* lat           = (const float*)d_in[2];  // [91]
    const float* lon           = (const float*)d_in[3];  // [180]
    const float* W             = (const float*)d_in[4];  // [64][64]
    const float* b             = (const float*)d_in[5];  // [64]
    float*       out           = (float*)d_out;          // [4][64][91][180]

    int*   knn_idx = (int*)d_ws;
    float* knn_w   = (float*)((char*)d_ws + (size_t)GRID * KNN_K * sizeof(int));

    knn_softmax_kernel<<<GRID, 32, 0, stream>>>(mesh_vertices, lat, lon, knn_idx, knn_w);
    interp_gemm_kernel<<<ROW_TILES, 32, 0, stream>>>(mesh_output, W, b, knn_idx, knn_w, out);
}